// _MultiHeadAttention_20864951124254
// MI455X (gfx1250) — compile-verified
//
#include <hip/hip_runtime.h>
#include <math.h>

typedef __attribute__((ext_vector_type(2))) float v2f;
typedef __attribute__((ext_vector_type(4))) float v4f;
typedef __attribute__((ext_vector_type(8))) float v8f;

#define B_SZ   4
#define SEQ    1024
#define DMODEL 1024
#define NH     16
#define DK     64
#define DV     64

// ---- f32 WMMA: D(16x16) = A(16x4) * B(4x16) + C ----
__device__ __forceinline__ v8f wmma4(v2f a, v2f b, v8f c) {
  // (neg_a, A, neg_b, B, c_mod, C, reuse_a, reuse_b)
  return __builtin_amdgcn_wmma_f32_16x16x4_f32(false, a, false, b, (short)0, c,
                                               false, false);
}

// =====================================================================
// Kernel 1: QKV projection.  out[p] = x[p] @ w[p] + bias[p].
// Q,K written [B,H,S,64]; V written TRANSPOSED [B,H,64,S] so the
// attention kernel's P@V B-fragments are contiguous b64 loads.
// 64x64 workgroup tile, 8 waves, each owning a 16(M) x 32(N) sub-tile.
// A tile: stride 20 (16B-aligned rows, b128 staging, conflict-free b64
// column fragment reads).  B tile transposed [n][k], stride 18.
// =====================================================================
__global__ __launch_bounds__(256) void qkv_proj_kernel(
    const float* __restrict__ qx, const float* __restrict__ kx, const float* __restrict__ vx,
    const float* __restrict__ wq, const float* __restrict__ wk, const float* __restrict__ wv,
    const float* __restrict__ bq, const float* __restrict__ bk, const float* __restrict__ bv,
    float* __restrict__ Qs, float* __restrict__ Ks, float* __restrict__ Vs) {
  const int p = blockIdx.z;
  const float* x    = (p == 0) ? qx : (p == 1) ? kx : vx;
  const float* w    = (p == 0) ? wq : (p == 1) ? wk : wv;
  const float* bias = (p == 0) ? bq : (p == 1) ? bk : bv;
  float*       o    = (p == 0) ? Qs : (p == 1) ? Ks : Vs;

  __shared__ float As[64][20];  // [m][k]; 80B rows: b128 stores + conflict-free b64 reads
  __shared__ float Bt[64][18];  // B tile TRANSPOSED, [n][k]

  const int t    = threadIdx.x;
  const int wid  = t >> 5;
  const int lane = t & 31;
  const int half = lane >> 4;
  const int ln   = lane & 15;
  const int m0   = (wid & 3) * 16;   // 4 M sub-tiles
  const int n0   = (wid >> 2) * 32;  // 2 N halves
  const int rowBase = blockIdx.y * 64;  // over M = B*S = 4096
  const int colBase = blockIdx.x * 64;  // over N = 1024

  // staging coordinates (constant across k-chunks)
  const int ar = t >> 2, ac = (t & 3) * 4;    // A: 64 rows x 16 cols, 1 float4/thread
  const int br = t >> 4, bc = (t & 15) * 4;   // B: 16 rows(k) x 64 cols(n), 1 float4/thread

  v8f acc0 = {};
  v8f acc1 = {};

  for (int k0 = 0; k0 < DMODEL; k0 += 16) {
    __syncthreads();
    // A tile: one b128 load + one b128 LDS store per thread
    *(v4f*)&As[ar][ac] =
        *(const v4f*)&x[(size_t)(rowBase + ar) * DMODEL + k0 + ac];
    // B tile: one b128 load + 4-way transpose scatter into Bt[n][k]
    {
      v4f wv4 = *(const v4f*)&w[(size_t)(k0 + br) * (NH * DK) + colBase + bc];
      Bt[bc + 0][br] = wv4.x;
      Bt[bc + 1][br] = wv4.y;
      Bt[bc + 2][br] = wv4.z;
      Bt[bc + 3][br] = wv4.w;
    }
    __syncthreads();
#pragma unroll
    for (int kk = 0; kk < 16; kk += 4) {
      // A frag (16x4): lane = M row; v0/v1 = K{0,1} (lanes 0-15), K{2,3} (lanes 16-31)
      v2f a  = *(const v2f*)&As[m0 + ln][kk + 2 * half];
      // B frag (4x16): col = n (ln), rows = k + 2*half, +1  -> contiguous in Bt[n][k]
      v2f b0 = *(const v2f*)&Bt[n0 + ln][kk + 2 * half];
      v2f b1 = *(const v2f*)&Bt[n0 + 16 + ln][kk + 2 * half];
      acc0 = wmma4(a, b0, acc0);
      acc1 = wmma4(a, b1, acc1);
    }
  }

  const int c0 = colBase + n0 + ln;
  const int c1 = c0 + 16;
  const float bv0 = bias[c0], bv1 = bias[c1];
  const int h0 = c0 >> 6, d0 = c0 & 63;
  const int h1 = c1 >> 6, d1 = c1 & 63;
#pragma unroll
  for (int r = 0; r < 8; r++) {  // C/D frag: VGPR r -> M = r + 8*half
    int grow = rowBase + m0 + r + 8 * half;  // 0..4095 over B*S
    int b = grow >> 10, s = grow & 1023;
    if (p == 2) {
      // V: transposed [B,H,D,S]
      o[(((size_t)b * NH + h0) * DV + d0) * SEQ + s] = acc0[r] + bv0;
      o[(((size_t)b * NH + h1) * DV + d1) * SEQ + s] = acc1[r] + bv1;
    } else {
      // Q,K: [B,H,S,D]
      o[(((size_t)b * NH + h0) * SEQ + s) * DK + d0] = acc0[r] + bv0;
      o[(((size_t)b * NH + h1) * SEQ + s) * DK + d1] = acc1[r] + bv1;
    }
  }
}

// =====================================================================
// Kernel 2: fused attention for one (b,h, 32-row query tile).
// scores (32x1024) live in LDS (~137 KB dynamic) -> softmax in LDS ->
// write attn + P@V, all without spilling scores to HBM.
// =====================================================================
#define QT   32
#define SROW 1028  // 1024 + 4 pad: column reads hit distinct banks; 16B-aligned rows
#define QSTR 68    // 64 + 4 pad; 272B row stride (16B aligned)

__global__ __launch_bounds__(256) void attn_kernel(
    const float* __restrict__ Qs, const float* __restrict__ Ks,
    const float* __restrict__ Vt, float* __restrict__ ctx,
    float* __restrict__ attn) {
  extern __shared__ float smem[];
  float* sc = smem;                     // [QT][SROW] scores / probs
  float* qt = smem + (size_t)QT * SROW; // [QT][QSTR] Q tile

  const int t    = threadIdx.x;
  const int wid  = t >> 5;
  const int lane = t & 31;
  const int half = lane >> 4;
  const int ln   = lane & 15;
  const int bh    = blockIdx.y;        // b*NH + h
  const int qbase = blockIdx.x * QT;
  const int b = bh >> 4, h = bh & 15;

  const float* Qp  = Qs + (size_t)bh * SEQ * DK;
  const float* Kp  = Ks + (size_t)bh * SEQ * DK;
  const float* Vtp = Vt + (size_t)bh * SEQ * DV;  // [DV][SEQ]
  float* attn_p = attn + (size_t)bh * SEQ * SEQ + (size_t)qbase * SEQ;

  // ---- load Q tile (32x64) into LDS, b128-wide ----
#pragma unroll
  for (int i = 0; i < 2; i++) {
    int idx4 = t + i * 256;                 // 512 float4 total
    int r = idx4 >> 4, c = (idx4 & 15) * 4;
    *(v4f*)&qt[r * QSTR + c] = *(const v4f*)&Qp[(size_t)(qbase + r) * DK + c];
  }
  __syncthreads();

  // ---- phase 1: scores = Q K^T / sqrt(dk) ----
  const float scale = 0.125f;  // 1/sqrt(64)
  for (int j = 0; j < 8; j++) {
    int kt = j * 8 + wid;  // key tile 0..63 (16 keys each)
    int n0 = kt * 16;
    v8f acc0 = {};
    v8f acc1 = {};
#pragma unroll
    for (int k4 = 0; k4 < 16; k4++) {
      int k0 = k4 * 4;
      v2f a0 = *(const v2f*)&qt[(0 + ln) * QSTR + k0 + 2 * half];
      v2f a1 = *(const v2f*)&qt[(16 + ln) * QSTR + k0 + 2 * half];
      // B = K^T fragment: col = key (ln), rows = dims k0+2*half, +1 -> one b64 / lane
      v2f bf = *(const v2f*)&Kp[(size_t)(n0 + ln) * DK + k0 + 2 * half];
      acc0 = wmma4(a0, bf, acc0);
      acc1 = wmma4(a1, bf, acc1);
    }
#pragma unroll
    for (int r = 0; r < 8; r++) {
      sc[(r + 8 * half) * SROW + n0 + ln]      = acc0[r] * scale;
      sc[(16 + r + 8 * half) * SROW + n0 + ln] = acc1[r] * scale;
    }
  }
  __syncthreads();

  // ---- phase 2: softmax, one wave32 per row ----
  for (int rr = 0; rr < 4; rr++) {
    float* rp = sc + (wid + rr * 8) * SROW;
    float m = -3.402823466e38f;
    for (int c = lane; c < SEQ; c += 32) m = fmaxf(m, rp[c]);
#pragma unroll
    for (int off = 16; off; off >>= 1) m = fmaxf(m, __shfl_xor(m, off, 32));
    float s = 0.f;
    for (int c = lane; c < SEQ; c += 32) {
      float e = __expf(rp[c] - m);
      rp[c] = e;
      s += e;
    }
#pragma unroll
    for (int off = 16; off; off >>= 1) s += __shfl_xor(s, off, 32);
    float inv = 1.0f / s;
    for (int c = lane; c < SEQ; c += 32) rp[c] *= inv;
  }
  __syncthreads();

  // ---- write attn tile: dominant HBM stream, b128-wide & fully coalesced ----
#pragma unroll 4
  for (int i = 0; i < 32; i++) {
    int idx4 = t + i * 256;                 // 8192 float4 = 32x1024 floats
    int r = idx4 >> 8, c = (idx4 & 255) * 4;
    *(v4f*)&attn_p[(size_t)r * SEQ + c] = *(const v4f*)&sc[r * SROW + c];
  }

  // ---- phase 3: context = P @ V ; 8 waves = 2(M) x 4(N) 16x16 tiles ----
  // V is [D][S]: fragment pairs along the key dim are contiguous -> b64 loads.
  const int m0 = (wid & 1) * 16;
  const int n0 = (wid >> 1) * 16;
  v8f acc = {};
#pragma unroll 4
  for (int k4 = 0; k4 < 256; k4++) {
    int k0 = k4 * 4;
    v2f a  = *(const v2f*)&sc[(m0 + ln) * SROW + k0 + 2 * half];
    v2f bf = *(const v2f*)&Vtp[(size_t)(n0 + ln) * SEQ + k0 + 2 * half];
    acc = wmma4(a, bf, acc);
  }
#pragma unroll
  for (int r = 0; r < 8; r++) {
    int srow = qbase + m0 + r + 8 * half;
    ctx[((size_t)b * SEQ + srow) * (NH * DV) + h * DV + n0 + ln] = acc[r];
  }
}

// =====================================================================
extern "C" void kernel_launch(void* const* d_in, const int* in_sizes, int n_in,
                              void* d_out, int out_size, void* d_ws, size_t ws_size,
                              hipStream_t stream) {
  (void)in_sizes; (void)n_in; (void)out_size; (void)ws_size;
  const float* q  = (const float*)d_in[0];
  const float* k  = (const float*)d_in[1];
  const float* v  = (const float*)d_in[2];
  const float* wq = (const float*)d_in[3];
  const float* wk = (const float*)d_in[4];
  const float* wv = (const float*)d_in[5];
  const float* bq = (const float*)d_in[6];
  const float* bk = (const float*)d_in[7];
  const float* bv = (const float*)d_in[8];
  // d_in[9] = attn_mask (falsy scalar in reference -> no masking)

  const size_t headElems = (size_t)B_SZ * NH * SEQ * DK;  // 4.19M floats each
  float* Qs = (float*)d_ws;
  float* Ks = Qs + headElems;
  float* Vs = Ks + headElems;  // total 48 MB of workspace (V stored transposed)

  float* ctx    = (float*)d_out;                              // [B,S,NH*DV]
  float* attn_o = ctx + (size_t)B_SZ * SEQ * NH * DV;         // [B,NH,S,S]

  // Kernel 1: 3 projections, 64x64 tiles over [4096 x 1024]
  qkv_proj_kernel<<<dim3((NH * DK) / 64, (B_SZ * SEQ) / 64, 3), 256, 0, stream>>>(
      q, k, v, wq, wk, wv, bq, bk, bv, Qs, Ks, Vs);

  // Kernel 2: fused attention; ~137 KB dynamic LDS (CDNA5 WGP has 320 KB)
  const size_t smemBytes = (size_t)(QT * SROW + QT * QSTR) * sizeof(float);
  (void)hipFuncSetAttribute((const void*)attn_kernel,
                            hipFuncAttributeMaxDynamicSharedMemorySize,
                            (int)smemBytes);
  attn_kernel<<<dim3(SEQ / QT, B_SZ * NH), 256, smemBytes, stream>>>(
      Qs, Ks, Vs, ctx, attn_o);
}